// SparseConv2D_77171972374829
// MI455X (gfx1250) — compile-verified
//
#include <hip/hip_runtime.h>

typedef __attribute__((ext_vector_type(2))) float v2f;
typedef __attribute__((ext_vector_type(8))) float v8f;

#define NB 50       // output blocks per spatial dim (50*14 = 700)
#define Hh 702
#define Ww 702
#define Cc 8
#define Ff 8
#define HO 700
#define WO 700
#define KDIM 72     // 3*3*8 reduction dim
#define KSTEPS 18   // KDIM / 4
#define PITCH 260   // padded LDS plane pitch (16*16 + 4) for bank-conflict-free writes

__global__ __launch_bounds__(128)
void sbnet_conv_wmma(const float* __restrict__ in, const float* __restrict__ mask,
                     const float* __restrict__ wt, const float* __restrict__ bias,
                     float* __restrict__ out) {
  __shared__ float s_patch[Cc * PITCH];   // input patch, planar [c][y*16+x]
  __shared__ float s_B[KDIM * 16];        // weights B[k][n], n padded to 16
  __shared__ float s_red[128];
  __shared__ float s_flag;

  const int bid = blockIdx.x;
  const int bx  = bid % NB;
  const int by  = (bid / NB) % NB;
  const int b   = bid / (NB * NB);
  const int tid = threadIdx.x;
  const int y0  = by * 14;
  const int x0  = bx * 14;

  // ---- 1. pooled max of mask over the 16x16 window (reduce_window max) ----
  float mx = -3.0e38f;
  #pragma unroll
  for (int i = tid; i < 256; i += 128) {
    int yy = i >> 4, xx = i & 15;
    mx = fmaxf(mx, mask[((size_t)b * Hh + (y0 + yy)) * Ww + (x0 + xx)]);
  }
  s_red[tid] = mx;
  __syncthreads();
  for (int s = 64; s > 0; s >>= 1) {
    if (tid < s) s_red[tid] = fmaxf(s_red[tid], s_red[tid + s]);
    __syncthreads();
  }
  if (tid == 0) s_flag = (s_red[0] > 0.5f) ? 1.0f : 0.0f;

  // ---- 2. load 16x16x8 input patch -> LDS planar [c][y][x] ----
  // Each input row (16 px * 8 ch) is 128 contiguous floats -> fully coalesced.
  #pragma unroll
  for (int i = tid; i < 16 * 16 * Cc; i += 128) {
    int y = i >> 7;
    int rem = i & 127;
    int x = rem >> 3;
    int c = rem & 7;
    float v = in[(((size_t)b * Hh + (y0 + y)) * Ww + (x0 + x)) * Cc + c];
    s_patch[c * PITCH + y * 16 + x] = v;
  }

  // ---- 3. load weights HWIO -> B[k][n], zero-pad n=8..15 ----
  #pragma unroll
  for (int i = tid; i < KDIM * 16; i += 128) {
    int k = i >> 4, n = i & 15;
    s_B[i] = (n < Ff) ? wt[k * Ff + n] : 0.0f;
  }
  __syncthreads();

  // ---- 4. implicit GEMM: M=196 (13 tiles of 16), N=16 (8 valid), K=72 ----
  const int lane = tid & 31;
  const int wave = tid >> 5;
  const int half = lane >> 4;          // 0: lanes 0-15, 1: lanes 16-31
  const int n    = lane & 15;
  const float bias_n = (n < Ff) ? bias[n] : 0.0f;
  const float flag = s_flag;

  // B fragments are identical for every M-tile: hoist into registers once.
  // B 4x16 f32 layout (dual of A): lanes 0-15 hold rows k0,k0+1; lanes 16-31 rows k0+2,k0+3.
  v2f Bfrag[KSTEPS];
  #pragma unroll
  for (int ks = 0; ks < KSTEPS; ks++) {
    int kk = ks * 4 + 2 * half;
    Bfrag[ks].x = s_B[kk * 16 + n];
    Bfrag[ks].y = s_B[(kk + 1) * 16 + n];
  }

  for (int t = wave; t < 13; t += 4) {       // wave-uniform: EXEC stays all-ones
    int m = t * 16 + (lane & 15);            // A-matrix row for this lane
    bool mvalid = (m < 196);
    int mm = mvalid ? m : 0;
    int oy = mm / 14, ox = mm % 14;
    int pixbase = oy * 16 + ox;

    v8f acc = {0.f, 0.f, 0.f, 0.f, 0.f, 0.f, 0.f, 0.f};
    #pragma unroll
    for (int ks = 0; ks < KSTEPS; ks++) {
      // A 16x4 f32 layout: lanes 0-15 hold K=k0,k0+1; lanes 16-31 hold K=k0+2,k0+3
      int kk = ks * 4 + 2 * half;
      int k1 = kk + 1;
      int kh0 = kk / 24, kw0 = (kk >> 3) % 3, c0 = kk & 7;
      int kh1 = k1 / 24, kw1 = (k1 >> 3) % 3, c1 = k1 & 7;
      v2f a;
      a.x = mvalid ? s_patch[c0 * PITCH + pixbase + kh0 * 16 + kw0] : 0.0f;
      a.y = mvalid ? s_patch[c1 * PITCH + pixbase + kh1 * 16 + kw1] : 0.0f;
      acc = __builtin_amdgcn_wmma_f32_16x16x4_f32(
          /*neg_a=*/false, a, /*neg_b=*/false, Bfrag[ks],
          /*c_mod=*/(short)0, acc, /*reuse_a=*/false, /*reuse_b=*/false);
    }

    // C/D layout: VGPR v -> M = v + 8*half, N = lane&15
    #pragma unroll
    for (int v = 0; v < 8; v++) {
      int ms = t * 16 + v + 8 * half;
      if (ms < 196 && n < Ff) {
        int oys = ms / 14, oxs = ms % 14;
        out[(((size_t)b * HO + (y0 + oys)) * WO + (x0 + oxs)) * Ff + n] =
            flag * (acc[v] + bias_n);
      }
    }
  }
}

extern "C" void kernel_launch(void* const* d_in, const int* in_sizes, int n_in,
                              void* d_out, int out_size, void* d_ws, size_t ws_size,
                              hipStream_t stream) {
  const float* in   = (const float*)d_in[0];
  const float* mask = (const float*)d_in[1];
  const float* wt   = (const float*)d_in[2];
  const float* bias = (const float*)d_in[3];
  float* out = (float*)d_out;

  dim3 grid(8 * NB * NB);   // one workgroup per (batch, 14x14 output block)
  dim3 block(128);          // 4 waves of 32
  sbnet_conv_wmma<<<grid, block, 0, stream>>>(in, mask, wt, bias, out);
}